// MultivariatePolynomial_50620484551271
// MI455X (gfx1250) — compile-verified
//
#include <hip/hip_runtime.h>
#include <hip/hip_bf16.h>

typedef float v2f __attribute__((ext_vector_type(2)));
typedef float v8f __attribute__((ext_vector_type(8)));

#define STEPS 27            // K' = 9 j-groups * 12 (padded k) / 4 per WMMA
#define ROWSTRIDE 18        // LDS row stride in floats (conflict-free)
#define TILEF (16 * ROWSTRIDE)
#define WAVEF (2 * TILEF)

__launch_bounds__(256)
__global__ void poly_wmma_kernel(const float* __restrict__ coeffs,
                                 const float* __restrict__ points,
                                 float* __restrict__ out, int np) {
  __shared__ float lds[8 * WAVEF];

  const int tid  = threadIdx.x;
  const int wave = tid >> 5;
  const int lane = tid & 31;
  const int lm   = lane & 15;   // point-in-tile (A/D row group) / column for B
  const int hi   = lane >> 4;   // lane half selects K sub-slot / D row offset

  const int b = blockIdx.y;
  const float* cb = coeffs + (size_t)b * 729;
  const float* pb = points + (size_t)b * np * 3;
  float* ob       = out    + (size_t)b * np;

  // ---- Preload B fragments (per-batch constant): B[K'][i] = C[i,j,k],
  // K' = j*12 + k, lane layout: b.x -> K=4s+2*hi, b.y -> K=4s+2*hi+1, col i=lm.
  v2f bf[STEPS];
  #pragma unroll
  for (int s = 0; s < STEPS; ++s) {
    const int j = s / 3;
    #pragma unroll
    for (int v = 0; v < 2; ++v) {
      const int k = (s % 3) * 4 + 2 * hi + v;   // runtime only via hi
      float val = 0.f;
      if (lm < 9 && k < 9) val = cb[lm * 81 + j * 9 + k];
      bf[s][v] = val;
    }
  }

  const int npm1 = np - 1;
  for (int base = blockIdx.x * 256; base < np; base += gridDim.x * 256) {
    const int tbase = base + wave * 32;
    int pA = tbase + lm;       if (pA > npm1) pA = npm1;   // tile0 powers
    int pB = tbase + 16 + lm;  if (pB > npm1) pB = npm1;   // tile1 powers
    const int pC = tbase + lane;                            // output point
    int pCc = pC;              if (pCc > npm1) pCc = npm1;

    const float y0 = pb[(size_t)pA * 3 + 1];
    const float z0 = pb[(size_t)pA * 3 + 2];
    const float y1 = pb[(size_t)pB * 3 + 1];
    const float z1 = pb[(size_t)pB * 3 + 2];
    const float xv = pb[(size_t)pCc * 3 + 0];

    // Power tables; pzt padded with zeros for k = 9..11 (K' padding).
    float py0[9], py1[9], pzt0[12], pzt1[12];
    py0[0] = 1.f; py1[0] = 1.f; pzt0[0] = 1.f; pzt1[0] = 1.f;
    #pragma unroll
    for (int t = 1; t < 9; ++t) {
      py0[t] = py0[t - 1] * y0;  py1[t] = py1[t - 1] * y1;
      pzt0[t] = pzt0[t - 1] * z0; pzt1[t] = pzt1[t - 1] * z1;
    }
    #pragma unroll
    for (int t = 9; t < 12; ++t) { pzt0[t] = 0.f; pzt1[t] = 0.f; }

    // Pre-select the lane-half-shifted z entries so the hot loop is mul-only.
    float pzl0[6], pzl1[6];
    #pragma unroll
    for (int c = 0; c < 3; ++c)
      #pragma unroll
      for (int v = 0; v < 2; ++v) {
        const int idx = c * 4 + v;
        pzl0[c * 2 + v] = hi ? pzt0[idx + 2] : pzt0[idx];
        pzl1[c * 2 + v] = hi ? pzt1[idx + 2] : pzt1[idx];
      }

    v8f d0 = {0.f, 0.f, 0.f, 0.f, 0.f, 0.f, 0.f, 0.f};
    v8f d1 = d0;
    #pragma unroll
    for (int s = 0; s < STEPS; ++s) {
      const int j = s / 3, c = s % 3;       // compile-time after unroll
      v2f a0, a1;
      a0.x = py0[j] * pzl0[c * 2];  a0.y = py0[j] * pzl0[c * 2 + 1];
      a1.x = py1[j] * pzl1[c * 2];  a1.y = py1[j] * pzl1[c * 2 + 1];
      d0 = __builtin_amdgcn_wmma_f32_16x16x4_f32(false, a0, false, bf[s],
                                                 (short)0, d0, false, false);
      d1 = __builtin_amdgcn_wmma_f32_16x16x4_f32(false, a1, false, bf[s],
                                                 (short)0, d1, false, false);
    }

    // Stage D tiles in LDS (row = point, col = i) so each lane can read
    // its own point's 9 partials contiguously. Stride 18 -> conflict-free.
    float* lw = lds + wave * WAVEF;
    #pragma unroll
    for (int r = 0; r < 8; ++r) {
      const int row = r + 8 * hi;
      lw[row * ROWSTRIDE + lm]         = d0[r];
      lw[TILEF + row * ROWSTRIDE + lm] = d1[r];
    }
    __syncthreads();

    const float* lr = lw + hi * TILEF + lm * ROWSTRIDE;
    float acc = 0.f, xp = 1.f;
    #pragma unroll
    for (int i = 0; i < 9; ++i) { acc = fmaf(lr[i], xp, acc); xp *= xv; }
    if (pC < np) ob[pC] = acc;
    __syncthreads();

    // Speculative prefetch of next iteration's point data.
    const int nb = base + gridDim.x * 256;
    if (nb < np) {
      int pn = nb + wave * 32 + lane; if (pn > npm1) pn = npm1;
      __builtin_prefetch(&pb[(size_t)pn * 3], 0, 1);
    }
  }
}

extern "C" void kernel_launch(void* const* d_in, const int* in_sizes, int n_in,
                              void* d_out, int out_size, void* d_ws, size_t ws_size,
                              hipStream_t stream) {
  const float* coeffs = (const float*)d_in[0];
  const float* points = (const float*)d_in[1];
  float* out = (float*)d_out;

  const int batch = in_sizes[0] / 729;               // 9*9*9 coeffs per batch
  const int np    = in_sizes[1] / (batch * 3);       // points per batch

  int tiles = (np + 255) / 256;
  int gx = tiles < 128 ? tiles : 128;
  if (gx < 1) gx = 1;
  dim3 grid(gx, batch), block(256);
  hipLaunchKernelGGL(poly_wmma_kernel, grid, block, 0, stream,
                     coeffs, points, out, np);
}